// HeatmapLoss_5806795784233
// MI455X (gfx1250) — compile-verified
//
#include <hip/hip_runtime.h>

#ifndef __has_builtin
#define __has_builtin(x) 0
#endif

typedef __attribute__((ext_vector_type(2))) float v2f;
typedef __attribute__((ext_vector_type(8))) float v8f;

#define HAVE_WMMA_F32_16X16X4 __has_builtin(__builtin_amdgcn_wmma_f32_16x16x4_f32)

// Exact f32 wave32 sum-reduction using V_WMMA_F32_16X16X4_F32.
// A is a 16x4 f32 matrix striped per ISA layout: lane L (0-15) holds row M=L,
// VGPR0=K0,VGPR1=K1; lanes 16-31 hold K2,K3. We place each lane's partial in
// its first A register and zero in the second; with B = all-ones,
// D[m][n] = partial(lane m) + partial(lane m+16)  (identical across n).
// Lanes 0-15 hold rows 0-7 in d[0..7]; lanes 16-31 hold rows 8-15.
// Sum d[0..7] then add the opposite half via shfl_xor(16) -> full wave sum in
// every lane. Must be called with EXEC all-ones (no divergence).
__device__ __forceinline__ float wave_reduce_sum(float x) {
#if HAVE_WMMA_F32_16X16X4
    v2f a; a.x = x;    a.y = 0.0f;
    v2f b; b.x = 1.0f; b.y = 1.0f;
    v8f c = {};
    v8f d = __builtin_amdgcn_wmma_f32_16x16x4_f32(
        /*neg_a=*/false, a, /*neg_b=*/false, b,
        /*c_mod=*/(short)0, c, /*reuse_a=*/false, /*reuse_b=*/false);
    float s = d[0] + d[1] + d[2] + d[3] + d[4] + d[5] + d[6] + d[7];
    s += __shfl_xor(s, 16, 32);
    return s;
#else
    for (int off = 16; off > 0; off >>= 1) x += __shfl_xor(x, off, 32);
    return x;
#endif
}

// One block per (b,n) keypoint image: 128x128 f32 tile, 256 threads (8 waves).
// HBM-bound: each thread streams 64 elements as 16x float4 (global_load_b128).
// Gaussian GT is built separably: s_ex/s_dx2 (per-column) in LDS, ey/dy2
// (per-row) in registers.
__global__ __launch_bounds__(256) void heatmap_loss_per_kp_kernel(
    const float* __restrict__ pred,    // [BN, 128, 128]
    const float* __restrict__ labels,  // [BN, 2]  (x, y) in [-1, 1)
    const float* __restrict__ mask,    // [BN]
    float* __restrict__ per_bn)        // [BN] masked per-keypoint loss
{
    constexpr int   H      = 128;
    constexpr int   W      = 128;
    constexpr float STRIDE = 4.0f;          // L / W = 512 / 128
    constexpr float START  = 1.5f;          // stride/2 - 0.5
    constexpr float INV2S2 = 1.0f / 128.0f; // 1 / (2 * 8^2)
    constexpr float D2MAX  = 128.0f * 4.6052f; // expo > TRUNC  <=>  d2 > 2s^2*TRUNC

    __shared__ float s_ex[W];
    __shared__ float s_dx2[W];
    __shared__ float s_wsum[8];

    const int bn  = blockIdx.x;
    const int tid = threadIdx.x;

    // Center (same computation in all threads; cheap, no extra sync).
    const float cx = fminf(fmaxf((labels[bn * 2 + 0] * 0.5f + 0.5f) * 512.0f, 0.0f), 511.0f);
    const float cy = fminf(fmaxf((labels[bn * 2 + 1] * 0.5f + 0.5f) * 512.0f, 0.0f), 511.0f);

    if (tid < W) {
        const float dx  = (START + STRIDE * (float)tid) - cx;
        const float dx2 = dx * dx;
        s_dx2[tid] = dx2;
        s_ex[tid]  = __expf(-dx2 * INV2S2);
    }

    const int   row  = tid >> 1;         // 0..127
    const int   col0 = (tid & 1) * 64;   // 0 or 64
    const float dy   = (START + STRIDE * (float)row) - cy;
    const float dy2  = dy * dy;
    const float ey   = __expf(-dy2 * INV2S2);

    __syncthreads();

    const float* p = pred + (size_t)bn * (H * W) + row * W + col0;
    float acc = 0.0f;
#pragma unroll
    for (int i = 0; i < 16; ++i) {
        const float4 pv = *reinterpret_cast<const float4*>(p + 4 * i);
        const float4 xv = *reinterpret_cast<const float4*>(&s_ex[col0 + 4 * i]);
        const float4 dv = *reinterpret_cast<const float4*>(&s_dx2[col0 + 4 * i]);

        const float g0 = (dv.x + dy2 > D2MAX) ? 0.0f : xv.x * ey;
        const float g1 = (dv.y + dy2 > D2MAX) ? 0.0f : xv.y * ey;
        const float g2 = (dv.z + dy2 > D2MAX) ? 0.0f : xv.z * ey;
        const float g3 = (dv.w + dy2 > D2MAX) ? 0.0f : xv.w * ey;

        const float e0 = pv.x - g0;
        const float e1 = pv.y - g1;
        const float e2 = pv.z - g2;
        const float e3 = pv.w - g3;
        acc = fmaf(e0, e0, acc);
        acc = fmaf(e1, e1, acc);
        acc = fmaf(e2, e2, acc);
        acc = fmaf(e3, e3, acc);
    }

    // Wave-level reduction via WMMA, then 8 wave partials through LDS.
    const float wsum = wave_reduce_sum(acc);
    const int wave = tid >> 5;
    const int lane = tid & 31;
    if (lane == 0) s_wsum[wave] = wsum;
    __syncthreads();

    if (tid == 0) {
        float t = 0.0f;
#pragma unroll
        for (int i = 0; i < 8; ++i) t += s_wsum[i];
        const float valid = (mask[bn] > 0.0f) ? 1.0f : 0.0f;
        per_bn[bn] = t * (1.0f / (float)(H * W)) * valid;
    }
}

// Single-block finalize: sum masked losses and mask values, emit scalar.
__global__ __launch_bounds__(256) void heatmap_loss_finalize_kernel(
    const float* __restrict__ per_bn,
    const float* __restrict__ mask,
    float* __restrict__ out,
    int n)
{
    __shared__ float s_l[8];
    __shared__ float s_m[8];

    const int tid = threadIdx.x;
    float lsum = 0.0f, msum = 0.0f;
    for (int i = tid; i < n; i += 256) {
        lsum += per_bn[i];
        msum += mask[i];
    }

    const float lw = wave_reduce_sum(lsum);
    const float mw = wave_reduce_sum(msum);
    const int wave = tid >> 5;
    const int lane = tid & 31;
    if (lane == 0) { s_l[wave] = lw; s_m[wave] = mw; }
    __syncthreads();

    if (tid == 0) {
        float lt = 0.0f, mt = 0.0f;
#pragma unroll
        for (int i = 0; i < 8; ++i) { lt += s_l[i]; mt += s_m[i]; }
        out[0] = (mt > 0.0f) ? (lt / fmaxf(mt, 1.0f)) : 0.0f;
    }
}

extern "C" void kernel_launch(void* const* d_in, const int* in_sizes, int n_in,
                              void* d_out, int out_size, void* d_ws, size_t ws_size,
                              hipStream_t stream) {
    const float* pred   = (const float*)d_in[0]; // [B,N,128,128] f32
    const float* labels = (const float*)d_in[1]; // [B,N,2] f32
    const float* mask   = (const float*)d_in[2]; // [B,N] f32
    float*       out    = (float*)d_out;         // scalar f32
    float*       per_bn = (float*)d_ws;          // BN floats of scratch

    const int BN = in_sizes[2]; // B*N = 544

    heatmap_loss_per_kp_kernel<<<BN, 256, 0, stream>>>(pred, labels, mask, per_bn);
    heatmap_loss_finalize_kernel<<<1, 256, 0, stream>>>(per_bn, mask, out, BN);
}